// Kernel_1105148710902228393_71605694759415
// MI455X (gfx1250) — compile-verified
//
#include <hip/hip_runtime.h>

// CDNA5 / gfx1250, wave32.
typedef __attribute__((ext_vector_type(2))) float v2f;
typedef __attribute__((ext_vector_type(8))) float v8f;

#define Nn 32
#define Cc 64
#define Hh 128
#define Ww 128
#define KK 448        // C * 7 taps
#define KSTEPS 112    // KK / 4 (K=4 per f32 WMMA)
#define KHALF 56      // K-steps per half (A-fragment reload granularity)
#define XS_VALID 146  // wl = w + 9, w in [-9, 136]
#define XS_STR 66     // LDS stride in dwords over channel dim (64 + 2 pad; bank step 2)
#define WPACK_FLOATS (4 * KSTEPS * 32 * 2)   // 28672 floats = 114,688 B

// ---- One-time weight repack into the exact WMMA A-fragment register image ----
// Wpack[mt][kk][lane] = { Wc[m][c*7+t], Wc[m][(c+1)*7+t] }
//   m = mt*16 + (lane&15), khalf = 2*(lane>>4), K-order k' = t*64 + c:
//   t = kk>>4, c = (4kk & 63) + khalf.
__global__ __launch_bounds__(256)
void Kernel_pack_weights(const float* __restrict__ Wc, float* __restrict__ wpack)
{
    const int p = blockIdx.x * 256 + threadIdx.x;        // pair index, 14336 total
    if (p >= 4 * KSTEPS * 32) return;
    const int lane = p & 31;
    const int kk   = (p >> 5) % KSTEPS;
    const int mt   = (p >> 5) / KSTEPS;
    const int m     = mt * 16 + (lane & 15);
    const int khalf = (lane >> 4) << 1;
    const int t     = kk >> 4;
    const int c     = ((kk << 2) & 63) + khalf;
    const float* wrow = Wc + (size_t)m * KK;
    v2f a;
    a.x = wrow[c * 7 + t];
    a.y = wrow[(c + 1) * 7 + t];
    *(v2f*)(wpack + ((size_t)p << 1)) = a;
}

// One workgroup = one (n, h) row. 8 waves:
//   mt = wave>>1 (out-channel tile, 16 ch), nt = (wave&1)*4 + j, j=0..3 (w tile, 16 w).
// t3 tile = Wconv(16x448) x Xcols(448x16) via V_WMMA_F32_16X16X4_F32.
// LDS holds x transposed as xs[wl][c]; interior staged with async global->LDS
// (per-lane LDS address does the transpose), halo zero-filled with DS stores.
template <bool PACKED>
__global__ __launch_bounds__(256)
void Kernel_fused_dilconv_roll(const float* __restrict__ x,
                               const float* __restrict__ Wc,    // (64,64,1,7): o*448 + c*7 + t
                               const float* __restrict__ wpack, // repacked A fragments
                               const float* __restrict__ p4w,   // 3 floats
                               float* __restrict__ out)
{
    __shared__ float xs[XS_VALID * XS_STR];   // 146*66*4 = 38,544 B

    const int tid  = threadIdx.x;
    const int lane = tid & 31;        // wave32
    const int wave = tid >> 5;
    const int blk  = blockIdx.x;
    const int n    = blk >> 7;        // H = 128
    const int h    = blk & (Hh - 1);

    const float p0 = p4w[0], p1 = p4w[1], p2 = p4w[2];

    // ---- Zero the w-halo: wl in {0..8} u {137..145}, all 64 channels ----
    for (int i = tid; i < 18 * Cc; i += 256) {
        const int c   = i & 63;
        const int idx = i >> 6;                   // 0..17
        const int wl  = (idx < 9) ? idx : (idx + 128);
        xs[wl * XS_STR + c] = 0.0f;
    }

    // ---- Stage interior via async global->LDS (ASYNCcnt), transposed [wl][c] ----
    // 64*128 elements / 256 threads = 32 uniform iterations; lanes -> consecutive w
    // (coalesced 128B global requests); per-lane LDS address handles the transpose.
    const float* xrow = x + ((size_t)(n * Cc) * Hh + h) * (size_t)Ww;
    for (int i = tid; i < Cc * Ww; i += 256) {
        const int c = i >> 7;                     // channel
        const int w = i & (Ww - 1);               // width
        const unsigned goff = (unsigned)(((c << 7) << 7) + (c ? 0 : 0)); // placeholder (see below)
        const unsigned gbyte = (unsigned)((c * (Hh * Ww) + w) * 4);      // byte offset from xrow
        const unsigned lbyte = (unsigned)(((w + 9) * XS_STR + c) * 4);   // LDS byte address
        (void)goff;
        asm volatile("global_load_async_to_lds_b32 %0, %1, %2"
                     :
                     : "v"(lbyte), "v"(gbyte), "s"(xrow)
                     : "memory");
    }
    asm volatile("s_wait_asynccnt 0x0" ::: "memory");   // own wave's async loads done
    __syncthreads();                                    // publish all waves' LDS writes

    const int mt    = wave >> 1;
    const int khalf = (lane >> 4) << 1;           // lanes>=16 hold K+2 (0 or 2)

    v8f acc[4];
#pragma unroll
    for (int j = 0; j < 4; ++j) acc[j] = (v8f){0.f,0.f,0.f,0.f,0.f,0.f,0.f,0.f};

#pragma unroll
    for (int half = 0; half < 2; ++half) {
        // A fragments for this K half: 56 x v2f = 112 VGPRs (total stays < 256)
        v2f afrag[KHALF];
        if (PACKED) {
            // coalesced global_load_b64 straight into even pairs, immediate offsets
            const float* wp = wpack +
                (((size_t)(mt * KSTEPS + half * KHALF) * 32 + lane) << 1);
#pragma unroll
            for (int q = 0; q < KHALF; ++q)
                afrag[q] = *(const v2f*)(wp + ((size_t)q << 6));  // q*32 pairs = 256 B
        } else {
            const int m = mt * 16 + (lane & 15);
            const float* wrow = Wc + (size_t)m * KK;
#pragma unroll
            for (int q = 0; q < KHALF; ++q) {
                const int kk = half * KHALF + q;
                const int t  = kk >> 4;
                const int c  = ((kk << 2) & 63) + khalf;
                v2f a;
                a.x = wrow[c * 7 + t];
                a.y = wrow[(c + 1) * 7 + t];
                afrag[q] = a;
            }
        }

#pragma unroll
        for (int j = 0; j < 4; ++j) {
            const int nt    = (wave & 1) * 4 + j;
            const int wbase = nt * 16 + (lane & 15);   // N index = w position
            const float* bbase = &xs[wbase * XS_STR + khalf];
#pragma unroll
            for (int q = 0; q < KHALF; ++q) {
                const int kk   = half * KHALF + q;
                const int t    = kk >> 4;
                const int coff = (kk << 2) & 63;
                // B[k'][w] = x[c][w + 3t - 9] -> xs[(wbase+3t)*66 + c]
                const v2f b = *(const v2f*)(bbase + (3 * t) * XS_STR + coff);
                acc[j] = __builtin_amdgcn_wmma_f32_16x16x4_f32(
                    false, afrag[q], false, b, (short)0, acc[j], false, false);
            }
        }
    }

    // ---- Fused epilogue: out = (x + t3) * t4 ----
    // C/D 16x16 f32 layout: VGPR v, lanes 0-15 -> M=v, lanes 16-31 -> M=v+8; N = lane&15.
    const int hprev = (h - 1) & (Hh - 1);              // roll(+1) on h, circular
#pragma unroll
    for (int j = 0; j < 4; ++j) {
        const int nt   = (wave & 1) * 4 + j;
        const int wcol = nt * 16 + (lane & 15);
        const int wp   = (wcol + 2) & (Ww - 1);        // roll(-2) on w, circular
#pragma unroll
        for (int v = 0; v < 8; ++v) {
            const int ch = mt * 16 + v + ((lane >> 4) << 3);
            const float* xch = x + (size_t)(n * Cc + ch) * (size_t)(Hh * Ww);
            // t4 taps: rows hprev-2, hprev, hprev+2 (zero outside), col wp
            const int r0 = hprev - 2, r2 = hprev + 2;
            const float v0 = (r0 >= 0) ? xch[r0 * Ww + wp] : 0.0f;
            const float v1 = xch[hprev * Ww + wp];
            const float v2 = (r2 < Hh) ? xch[r2 * Ww + wp] : 0.0f;
            const float t4 = p0 * v0 + p1 * v1 + p2 * v2;
            const float xv = xs[(wcol + 9) * XS_STR + ch];   // x[n,ch,h,wcol]
            out[((size_t)(n * Cc + ch) * Hh + h) * Ww + wcol] = (xv + acc[j][v]) * t4;
        }
    }
}

extern "C" void kernel_launch(void* const* d_in, const int* in_sizes, int n_in,
                              void* d_out, int out_size, void* d_ws, size_t ws_size,
                              hipStream_t stream) {
    const float* x   = (const float*)d_in[0];   // (32, 64, 128, 128) fp32
    const float* Wc  = (const float*)d_in[1];   // (64, 64, 1, 7) fp32
    const float* p4w = (const float*)d_in[2];   // (3, 1) fp32
    float* out = (float*)d_out;                 // (32, 64, 128, 128) fp32
    float* wpack = (float*)d_ws;

    dim3 grid(Nn * Hh);                         // 4096 workgroups, one per (n, h) row
    dim3 block(256);                            // 8 waves (wave32)

    if (ws_size >= (size_t)WPACK_FLOATS * sizeof(float)) {
        // Repack weights into the A-fragment register image, then fused GEMM.
        Kernel_pack_weights<<<dim3((4 * KSTEPS * 32 + 255) / 256), block, 0, stream>>>(Wc, wpack);
        Kernel_fused_dilconv_roll<true><<<grid, block, 0, stream>>>(x, Wc, wpack, p4w, out);
    } else {
        Kernel_fused_dilconv_roll<false><<<grid, block, 0, stream>>>(x, Wc, wpack, p4w, out);
    }
}